// ContrastiveProtosTeacherLoss_82712480186748
// MI455X (gfx1250) — compile-verified
//
#include <hip/hip_runtime.h>
#include <math.h>

#define V_ 2
#define C_ 64
#define H_ 96
#define W_ 96
#define D_ 32
#define N_ 2304
#define P_ 144
#define CH_STRIDE (H_ * W_ * D_) /* 294912 */

__constant__ const float kNothing = 0.0f; // keep TU non-trivial

constexpr float EPSF = 1e-8f;
constexpr float SIG2 = (128.0f / 2.355f) * (128.0f / 2.355f);
constexpr float INV2SIG2 = 1.0f / (2.0f * SIG2);
constexpr float INV_TEMP_T = 1.0f / 0.015f;
constexpr float INV_TEMP_S = 1.0f / 0.03f;

typedef float v2f __attribute__((ext_vector_type(2)));
typedef float v8f __attribute__((ext_vector_type(8)));

// ---------------------------------------------------------------------------
// Reflect-pad unnormalize (exact port of reference `unnorm`)
// ---------------------------------------------------------------------------
__device__ __forceinline__ float unnorm_reflect(float c, float size) {
  float x = ((c + 1.0f) * size - 1.0f) * 0.5f;
  float xr = fabsf(x + 0.5f);
  float extra = fmodf(xr, size);
  float flips = floorf(xr / size);
  float m = fmodf(flips, 2.0f);
  x = (m == 0.0f) ? extra : (size - extra);
  x -= 0.5f;
  return fminf(fmaxf(x, 0.0f), size - 1.0f);
}

// ---------------------------------------------------------------------------
// Grid-sample kernel: one block (64 threads) per sample point.
// emb layout (B,C,H,W,D); logical transposed view (B,C,D,H,W) is sampled.
// Outputs: out_norm (v,n,64) l2-normalized, optional out_raw, coords (v,n,3).
// ---------------------------------------------------------------------------
__global__ __launch_bounds__(64) void sample_kernel(
    const float* __restrict__ emb, const float* __restrict__ coord,
    const float* __restrict__ jit, int Dr, int Hr, int Wr, float sd, float sh,
    float sw, int nTot, float* __restrict__ out_norm,
    float* __restrict__ out_raw, float* __restrict__ out_coords) {
  const int n = blockIdx.x;
  const int v = blockIdx.y;
  const int c = threadIdx.x;
  const int ix = n % Wr;
  const int iy = (n / Wr) % Hr;
  const int iz = n / (Wr * Hr);

  // jitter order (d,h,w); grid channels are (x,y,z) = reversed
  const float jz = (jit[v * 3 + 0] - 0.5f) * 2.0f * sd;
  const float jy = (jit[v * 3 + 1] - 0.5f) * 2.0f * sh;
  const float jx = (jit[v * 3 + 2] - 0.5f) * 2.0f * sw;
  const float gx = ((float)ix + 0.5f) * 2.0f / (float)Wr - 1.0f + jx;
  const float gy = ((float)iy + 0.5f) * 2.0f / (float)Hr - 1.0f + jy;
  const float gz = ((float)iz + 0.5f) * 2.0f / (float)Dr - 1.0f + jz;

  const float x = unnorm_reflect(gx, (float)W_);
  const float y = unnorm_reflect(gy, (float)H_);
  const float z = unnorm_reflect(gz, (float)D_);
  const float x0f = floorf(x), y0f = floorf(y), z0f = floorf(z);
  const float fx = x - x0f, fy = y - y0f, fz = z - z0f;
  int x0 = max(min((int)x0f, W_ - 1), 0);
  int y0 = max(min((int)y0f, H_ - 1), 0);
  int z0 = max(min((int)z0f, D_ - 1), 0);
  const int x1 = min(x0 + 1, W_ - 1);
  const int y1 = min(y0 + 1, H_ - 1);
  const int z1 = min(z0 + 1, D_ - 1);
  const float wx0 = 1.0f - fx, wx1 = fx;
  const float wy0 = 1.0f - fy, wy1 = fy;
  const float wz0 = 1.0f - fz, wz1 = fz;

  // original layout strides: h: W*D, w: D, d: 1
  const size_t oy0 = (size_t)y0 * (W_ * D_), oy1 = (size_t)y1 * (W_ * D_);
  const size_t ox0 = (size_t)x0 * D_, ox1 = (size_t)x1 * D_;

  auto tri = [&](const float* bp) -> float {
    float v00 = wz0 * bp[oy0 + ox0 + z0] + wz1 * bp[oy0 + ox0 + z1];
    float v01 = wz0 * bp[oy0 + ox1 + z0] + wz1 * bp[oy0 + ox1 + z1];
    float v10 = wz0 * bp[oy1 + ox0 + z0] + wz1 * bp[oy1 + ox0 + z1];
    float v11 = wz0 * bp[oy1 + ox1 + z0] + wz1 * bp[oy1 + ox1 + z1];
    return wy0 * (wx0 * v00 + wx1 * v01) + wy1 * (wx0 * v10 + wx1 * v11);
  };

  const float* ebase = emb + ((size_t)v * C_ + c) * CH_STRIDE;
  float val = tri(ebase);

  __shared__ float red[64];
  red[c] = val * val;
  __syncthreads();
  for (int s = 32; s > 0; s >>= 1) {
    if (c < s) red[c] += red[c + s];
    __syncthreads();
  }
  const float nrm = fmaxf(sqrtf(red[0]), EPSF);

  const size_t oo = ((size_t)v * nTot + n) * C_ + c;
  out_norm[oo] = val / nrm;
  if (out_raw) out_raw[oo] = val;
  if (c < 3) {
    const float* cbase = coord + ((size_t)v * 3 + c) * CH_STRIDE;
    out_coords[((size_t)v * nTot + n) * 3 + c] = tri(cbase);
  }
}

// ---------------------------------------------------------------------------
// WMMA fp32 matmul: out(v, N, 144) = A(v, N, 64) @ B(v, 144, 64)^T
// One wave per 16-row strip; 9 column tiles held in accumulators.
// Fragment layouts per CDNA5 ISA 7.12.2 (32-bit A 16x4, C/D 16x16).
// ---------------------------------------------------------------------------
__global__ __launch_bounds__(32) void simtile_wmma(
    const float* __restrict__ A, const float* __restrict__ Bm,
    float* __restrict__ out, int N) {
  const int v = blockIdx.y;
  const int mt = blockIdx.x;  // row tile (16 rows)
  const int lane = threadIdx.x;
  const int lid = lane & 15;
  const int khalf = (lane >> 4) * 2;  // K offset of this lane half

  const float* Abase = A + ((size_t)v * N + (size_t)mt * 16 + lid) * C_ + khalf;
  const float* Bbase = Bm + ((size_t)v * P_ + lid) * C_ + khalf;

  v8f acc[9];
#pragma unroll
  for (int j = 0; j < 9; j++) {
    v8f z = {0.f, 0.f, 0.f, 0.f, 0.f, 0.f, 0.f, 0.f};
    acc[j] = z;
  }

#pragma unroll
  for (int kk = 0; kk < 16; kk++) {
    v2f a;
    a.x = Abase[kk * 4 + 0];
    a.y = Abase[kk * 4 + 1];
#pragma unroll
    for (int j = 0; j < 9; j++) {
      const float* bp = Bbase + (size_t)j * 16 * C_ + kk * 4;
      v2f b;
      b.x = bp[0];
      b.y = bp[1];
      acc[j] = __builtin_amdgcn_wmma_f32_16x16x4_f32(
          /*neg_a=*/false, a, /*neg_b=*/false, b,
          /*c_mod=*/(short)0, acc[j], /*reuse_a=*/false, /*reuse_b=*/false);
    }
  }

  const int rowbase = mt * 16 + (lane >> 4) * 8;  // M = r + 8*(lane>=16)
#pragma unroll
  for (int j = 0; j < 9; j++) {
#pragma unroll
    for (int r = 0; r < 8; r++) {
      out[((size_t)v * N + rowbase + r) * P_ + j * 16 + lid] = acc[j][r];
    }
  }
}

// ---------------------------------------------------------------------------
// Row softmax (T=0.015); mode 0: multiply by Gaussian spatial weight -> w;
// mode 1: plain softmax -> targets.
// ---------------------------------------------------------------------------
__global__ __launch_bounds__(256) void rowsoft_kernel(
    const float* __restrict__ simin, const float* __restrict__ coords_t,
    const float* __restrict__ coords_p, float* __restrict__ outbuf, int mode) {
  const int n = blockIdx.x;
  const int v = blockIdx.y;
  const int p = threadIdx.x;
  __shared__ float red[256];

  float xv = 0.0f;
  float xm = -3.4e38f;
  if (p < P_) {
    xv = simin[((size_t)v * N_ + n) * P_ + p] * INV_TEMP_T;
    xm = xv;
  }
  red[p] = xm;
  __syncthreads();
  for (int s = 128; s > 0; s >>= 1) {
    if (p < s) red[p] = fmaxf(red[p], red[p + s]);
    __syncthreads();
  }
  const float mx = red[0];
  __syncthreads();
  const float e = (p < P_) ? expf(xv - mx) : 0.0f;
  red[p] = e;
  __syncthreads();
  for (int s = 128; s > 0; s >>= 1) {
    if (p < s) red[p] += red[p + s];
    __syncthreads();
  }
  const float sm = red[0];

  if (p < P_) {
    float a = e / sm;
    if (mode == 0) {
      const float t0 = coords_t[((size_t)v * N_ + n) * 3 + 0];
      const float t1 = coords_t[((size_t)v * N_ + n) * 3 + 1];
      const float t2 = coords_t[((size_t)v * N_ + n) * 3 + 2];
      const float q0 = coords_p[((size_t)v * P_ + p) * 3 + 0];
      const float q1 = coords_p[((size_t)v * P_ + p) * 3 + 1];
      const float q2 = coords_p[((size_t)v * P_ + p) * 3 + 2];
      const float dx = t0 - q0, dy = t1 - q1, dz = (t2 - q2) * 2.0f;
      const float d2 = dx * dx + dy * dy + dz * dz;
      a *= expf(-d2 * INV2SIG2);
    }
    outbuf[((size_t)v * N_ + n) * P_ + p] = a;
  }
}

// ---------------------------------------------------------------------------
// Proto update: protos[p,c] = l2norm( (w^T t_emb)[p,:] / den ), coords update.
// One block per (v,p); thread c reduces over N (coalesced 256B row reads).
// ---------------------------------------------------------------------------
__global__ __launch_bounds__(64) void protoupd_kernel(
    const float* __restrict__ wbuf, const float* __restrict__ t_emb,
    const float* __restrict__ coords_t, float* __restrict__ protos_n,
    float* __restrict__ coords_p, int N) {
  const int p = blockIdx.x;
  const int v = blockIdx.y;
  const int c = threadIdx.x;

  float acc = 0.0f, den = 0.0f, cacc = 0.0f;
  for (int n = 0; n < N; n++) {
    const float w = wbuf[((size_t)v * N + n) * P_ + p];  // broadcast read
    acc += w * t_emb[((size_t)v * N + n) * C_ + c];
    den += w;
    if (c < 3) cacc += w * coords_t[((size_t)v * N + n) * 3 + c];
  }
  den += EPSF;
  const float val = acc / den;

  __shared__ float red[64];
  red[c] = val * val;
  __syncthreads();
  for (int s = 32; s > 0; s >>= 1) {
    if (c < s) red[c] += red[c + s];
    __syncthreads();
  }
  const float nrm = fmaxf(sqrtf(red[0]), EPSF);
  protos_n[((size_t)v * P_ + p) * C_ + c] = val / nrm;
  if (c < 3) coords_p[((size_t)v * P_ + p) * 3 + c] = cacc / den;
}

// ---------------------------------------------------------------------------
// Student CE: per (v,m) block — argmin over teacher coords, log-softmax of
// logits, masked target CE; atomic accumulate (ce*valid, valid).
// ---------------------------------------------------------------------------
__global__ __launch_bounds__(256) void student_kernel(
    const float* __restrict__ logits, const float* __restrict__ coords_s,
    const float* __restrict__ coords_t, const float* __restrict__ coords_p,
    const float* __restrict__ targets, float* __restrict__ accum, int N) {
  const int m = blockIdx.x;
  const int v = blockIdx.y;
  const int t = threadIdx.x;
  __shared__ float redf[256];
  __shared__ int redi[256];

  const float s0 = coords_s[((size_t)v * N + m) * 3 + 0];
  const float s1 = coords_s[((size_t)v * N + m) * 3 + 1];
  const float s2 = coords_s[((size_t)v * N + m) * 3 + 2];

  // --- nearest teacher point ---
  float bestd = 3.4e38f;
  int besti = 0x7fffffff;
  for (int n = t; n < N; n += 256) {
    const float* ct = &coords_t[((size_t)v * N + n) * 3];
    const float dx = s0 - ct[0], dy = s1 - ct[1], dz = (s2 - ct[2]) * 2.0f;
    const float d2 = dx * dx + dy * dy + dz * dz;
    if (d2 < bestd) {
      bestd = d2;
      besti = n;
    }
  }
  redf[t] = bestd;
  redi[t] = besti;
  __syncthreads();
  for (int s = 128; s > 0; s >>= 1) {
    if (t < s) {
      if (redf[t + s] < redf[t] ||
          (redf[t + s] == redf[t] && redi[t + s] < redi[t])) {
        redf[t] = redf[t + s];
        redi[t] = redi[t + s];
      }
    }
    __syncthreads();
  }
  const float dmin = redf[0];
  const int idx = redi[0];
  const float valid = (dmin <= 9.0f) ? 1.0f : 0.0f;  // sqrt(dmin) <= 3
  __syncthreads();

  // --- log-softmax of logits row ---
  float xv = 0.0f, xm = -3.4e38f;
  if (t < P_) {
    xv = logits[((size_t)v * N + m) * P_ + t] * INV_TEMP_S;
    xm = xv;
  }
  redf[t] = xm;
  __syncthreads();
  for (int s = 128; s > 0; s >>= 1) {
    if (t < s) redf[t] = fmaxf(redf[t], redf[t + s]);
    __syncthreads();
  }
  const float mx = redf[0];
  __syncthreads();
  const float e = (t < P_) ? expf(xv - mx) : 0.0f;
  redf[t] = e;
  __syncthreads();
  for (int s = 128; s > 0; s >>= 1) {
    if (t < s) redf[t] += redf[t + s];
    __syncthreads();
  }
  const float lse = logf(redf[0]);
  __syncthreads();

  // --- masked target ---
  float tg = 0.0f;
  if (t < P_) {
    const float* cp = &coords_p[((size_t)v * P_ + t) * 3];
    const float dx = s0 - cp[0], dy = s1 - cp[1], dz = (s2 - cp[2]) * 2.0f;
    const float d2 = dx * dx + dy * dy + dz * dz;
    const float g = expf(-d2 * INV2SIG2);
    const float mask = (g >= 0.5f) ? 1.0f : 0.0f;
    tg = targets[((size_t)v * N + idx) * P_ + t] * mask;
  }
  redf[t] = tg;
  __syncthreads();
  for (int s = 128; s > 0; s >>= 1) {
    if (t < s) redf[t] += redf[t + s];
    __syncthreads();
  }
  const float ts = redf[0];
  __syncthreads();

  float ce_part = 0.0f;
  if (t < P_) {
    const float tn = tg / (ts + EPSF);
    const float logp = (xv - mx) - lse;
    ce_part = -tn * logp;
  }
  redf[t] = ce_part;
  __syncthreads();
  for (int s = 128; s > 0; s >>= 1) {
    if (t < s) redf[t] += redf[t + s];
    __syncthreads();
  }
  if (t == 0) {
    atomicAdd(&accum[0], redf[0] * valid);
    atomicAdd(&accum[1], valid);
  }
}

__global__ void init_accum_kernel(float* a) {
  if (threadIdx.x < 4) a[threadIdx.x] = 0.0f;
}

__global__ void finalize_kernel(const float* __restrict__ accum,
                                float* __restrict__ out) {
  const float total =
      accum[0] / (accum[1] + EPSF) + accum[2] / (accum[3] + EPSF);
  out[0] = 1.0f * total / 2.0f;  // LOSS_WEIGHT * total / 2
}

// ---------------------------------------------------------------------------
extern "C" void kernel_launch(void* const* d_in, const int* in_sizes, int n_in,
                              void* d_out, int out_size, void* d_ws,
                              size_t ws_size, hipStream_t stream) {
  (void)in_sizes;
  (void)n_in;
  (void)out_size;
  (void)ws_size;
  const float* e_s0 = (const float*)d_in[0];
  const float* e_s1 = (const float*)d_in[1];
  const float* e_t = (const float*)d_in[2];
  const float* c_s0 = (const float*)d_in[3];
  const float* c_s1 = (const float*)d_in[4];
  const float* c_t = (const float*)d_in[5];
  const float* j_p = (const float*)d_in[7];
  const float* j_s0 = (const float*)d_in[8];
  const float* j_s1 = (const float*)d_in[9];
  const float* j_t = (const float*)d_in[10];

  float* ws = (float*)d_ws;
  size_t off = 0;
  auto alloc = [&](size_t n) {
    float* p = ws + off;
    off += n;
    return p;
  };
  float* protos_n = alloc((size_t)V_ * P_ * C_);
  float* coords_p = alloc((size_t)V_ * P_ * 3);
  float* t_emb = alloc((size_t)V_ * N_ * C_);
  float* t_emb_n = alloc((size_t)V_ * N_ * C_);
  float* coords_t = alloc((size_t)V_ * N_ * 3);
  float* s_emb_n0 = alloc((size_t)V_ * N_ * C_);
  float* coords_s0 = alloc((size_t)V_ * N_ * 3);
  float* s_emb_n1 = alloc((size_t)V_ * N_ * C_);
  float* coords_s1 = alloc((size_t)V_ * N_ * 3);
  float* simbuf = alloc((size_t)V_ * N_ * P_);  // also w-buffer / logits
  float* targets = alloc((size_t)V_ * N_ * P_);
  float* accum = alloc(4);

  const float sdP = 8.0f / 32.0f, shP = 16.0f / 96.0f, swP = 16.0f / 96.0f;
  const float sdS = 2.0f / 32.0f, shS = 8.0f / 96.0f, swS = 8.0f / 96.0f;

  // 1) grid samples
  sample_kernel<<<dim3(P_, V_), 64, 0, stream>>>(
      e_t, c_t, j_p, 4, 6, 6, sdP, shP, swP, P_, protos_n, nullptr, coords_p);
  sample_kernel<<<dim3(N_, V_), 64, 0, stream>>>(
      e_t, c_t, j_t, 16, 12, 12, sdS, shS, swS, N_, t_emb_n, t_emb, coords_t);
  sample_kernel<<<dim3(N_, V_), 64, 0, stream>>>(e_s0, c_s0, j_s0, 16, 12, 12,
                                                 sdS, shS, swS, N_, s_emb_n0,
                                                 nullptr, coords_s0);
  sample_kernel<<<dim3(N_, V_), 64, 0, stream>>>(e_s1, c_s1, j_s1, 16, 12, 12,
                                                 sdS, shS, swS, N_, s_emb_n1,
                                                 nullptr, coords_s1);
  init_accum_kernel<<<1, 4, 0, stream>>>(accum);

  // 2) soft k-means iterations
  for (int it = 0; it < 3; it++) {
    simtile_wmma<<<dim3(N_ / 16, V_), 32, 0, stream>>>(t_emb_n, protos_n,
                                                       simbuf, N_);
    rowsoft_kernel<<<dim3(N_, V_), 256, 0, stream>>>(simbuf, coords_t,
                                                     coords_p, simbuf, 0);
    protoupd_kernel<<<dim3(P_, V_), 64, 0, stream>>>(simbuf, t_emb, coords_t,
                                                     protos_n, coords_p, N_);
  }

  // 3) targets from final protos
  simtile_wmma<<<dim3(N_ / 16, V_), 32, 0, stream>>>(t_emb_n, protos_n, simbuf,
                                                     N_);
  rowsoft_kernel<<<dim3(N_, V_), 256, 0, stream>>>(simbuf, nullptr, nullptr,
                                                   targets, 1);

  // 4) students
  simtile_wmma<<<dim3(N_ / 16, V_), 32, 0, stream>>>(s_emb_n0, protos_n, simbuf,
                                                     N_);
  student_kernel<<<dim3(N_, V_), 256, 0, stream>>>(
      simbuf, coords_s0, coords_t, coords_p, targets, accum + 0, N_);

  simtile_wmma<<<dim3(N_ / 16, V_), 32, 0, stream>>>(s_emb_n1, protos_n, simbuf,
                                                     N_);
  student_kernel<<<dim3(N_, V_), 256, 0, stream>>>(
      simbuf, coords_s1, coords_t, coords_p, targets, accum + 2, N_);

  // 5) scalar loss
  finalize_kernel<<<1, 1, 0, stream>>>(accum, (float*)d_out);
}